// CategoricalLoss_78606491452302
// MI455X (gfx1250) — compile-verified
//
#include <hip/hip_runtime.h>
#include <math.h>
#include <limits.h>

// Problem constants from the reference: B=4, T=1024, V=32000, C=64
#define B_ 4
#define T_ 1024
#define V_ 32000
#define C_ 64
#define NROWS (B_ * T_)

typedef __attribute__((ext_vector_type(2))) float v2f;
typedef __attribute__((ext_vector_type(8))) float v8f;

// Online-softmax merge: combine (m,s,idx) running state. Deterministic order everywhere.
__device__ __forceinline__ void msi_combine(float& m, float& s, int& idx,
                                            float m2, float s2, int idx2) {
    if (m2 > m) {
        s = s * __expf(m - m2) + s2;
        m = m2;
        idx = idx2;
    } else if (m2 == m) {
        s += s2;
        idx = min(idx, idx2);   // jnp.argmax takes first max
    } else {
        s += s2 * __expf(m2 - m);
    }
}

// ---------------------------------------------------------------------------
// Kernel 1: one block per (b,t) row. Single streaming pass over 32000 logits:
// fused argmax + online log-sum-exp. Coalesced float4 (global_load_b128).
// ---------------------------------------------------------------------------
__global__ __launch_bounds__(256) void rowpass_kernel(
    const float* __restrict__ logits, const int* __restrict__ targets,
    int* __restrict__ gens, float* __restrict__ nllw, float* __restrict__ wsum) {
    const int row = blockIdx.x;                 // 0 .. B*T-1
    const int tid = threadIdx.x;
    const float4* p = (const float4*)(logits + (size_t)row * V_);

    float m = -INFINITY, s = 0.0f;
    int idx = INT_MAX;

    // V/4 = 8000 float4s, stride-256 across the block: fully coalesced 16B loads.
    for (int i = tid; i < V_ / 4; i += 256) {
        float4 v = p[i];
        const int j = i * 4;
        float xv[4] = {v.x, v.y, v.z, v.w};
#pragma unroll
        for (int k = 0; k < 4; ++k) {
            float x = xv[k];
            if (x > m) {                        // new max: rescale running sum
                s = s * __expf(m - x) + 1.0f;
                m = x;
                idx = j + k;
            } else {
                s += __expf(x - m);
            }
        }
    }

    // wave32 butterfly reduction
#pragma unroll
    for (int off = 16; off > 0; off >>= 1) {
        float m2 = __shfl_xor(m, off);
        float s2 = __shfl_xor(s, off);
        int i2 = __shfl_xor(idx, off);
        msi_combine(m, s, idx, m2, s2, i2);
    }

    // cross-wave merge (8 waves), sequential in thread 0 => deterministic
    __shared__ float sm[8], ss[8];
    __shared__ int si[8];
    const int wave = tid >> 5, lane = tid & 31;
    if (lane == 0) { sm[wave] = m; ss[wave] = s; si[wave] = idx; }
    __syncthreads();
    if (tid == 0) {
        for (int wv = 1; wv < 8; ++wv) msi_combine(m, s, idx, sm[wv], ss[wv], si[wv]);
        const int tgt = targets[row];
        const float xt = logits[(size_t)row * V_ + tgt];
        const float lse = m + logf(s);          // log-sum-exp of the row
        const float w = (tgt != 0) ? 1.0f : 0.0f;
        gens[row] = idx;                        // argmax token
        nllw[row] = w * (lse - xt);             // w * NLL
        wsum[row] = w;
    }
}

// ---------------------------------------------------------------------------
// Kernel 2: attnsum via V_WMMA_F32_16X16X4_F32 (full f32 precision).
// One wave per tile of 16 consecutive t. D[M,N] = sum_c sel[t0+M,c]*attns[c,t0+N];
// the diagonal D[M,M] is attnsum[t0+M].
// A layout (ISA 7.12.2, 32-bit A 16x4): lanes 0-15 = rows, VGPR0/1 = K0,K1;
//                                       lanes 16-31 = rows, VGPR0/1 = K2,K3.
// B/D layout: rows striped across lanes within a VGPR (half-split at lane 16).
// ---------------------------------------------------------------------------
__global__ __launch_bounds__(32) void attn_wmma_kernel(
    const float* __restrict__ attns, const unsigned char* __restrict__ assoc,
    const int* __restrict__ gens, float* __restrict__ attnterm) {
    const int tile = blockIdx.x;                // 0 .. B*T/16 - 1
    const int b = tile / (T_ / 16);
    const int t0 = (tile % (T_ / 16)) * 16;
    const int l = threadIdx.x;                  // 0..31, EXEC all ones for WMMA
    const int n = l & 15;                       // row / column within the tile
    const int khalf = l >> 4;                   // which K-pair this lane supplies

    const int g = gens[b * T_ + t0 + n];        // generated token of row M=n
    const unsigned char* arow = assoc + (size_t)g * C_;
    const float* acol = attns + (size_t)b * C_ * T_ + t0;   // attns[b,c,t] = [(b*C+c)*T + t]

    v8f acc = {};
    int anyv = 0;                               // "this lane saw an assoc bit" (rows l&15)
#pragma unroll
    for (int kc = 0; kc < 16; ++kc) {           // K = 64 in chunks of 4
        const int c0 = kc * 4 + khalf * 2;
        const unsigned char s0 = arow[c0], s1 = arow[c0 + 1];
        anyv |= (int)(s0 | s1);
        v2f a; a.x = s0 ? 1.0f : 0.0f; a.y = s1 ? 1.0f : 0.0f;
        v2f bb;
        bb.x = acol[(size_t)(c0)     * T_ + n];
        bb.y = acol[(size_t)(c0 + 1) * T_ + n];
        acc = __builtin_amdgcn_wmma_f32_16x16x4_f32(
            false, a, false, bb, (short)0, acc, false, false);
    }

    // has_assoc for row (l&15): lanes l and l^16 cover complementary categories
    const int anyRow = anyv | __shfl_xor(anyv, 16);

    // Diagonal of D: VGPR r holds M=r (lanes 0-15, N=lane) and M=r+8 (lanes 16-31).
    // => lane r (r<8) owns D[r,r] in acc[r]; lane r+24 owns D[r+8,r+8] in acc[r].
    float diag = 0.0f;
#pragma unroll
    for (int r = 0; r < 8; ++r)
        diag = ((l == r) || (l == r + 24)) ? acc[r] : diag;

    if (l < 8 || l >= 24) {
        const int M = (l < 8) ? l : (l - 16);
        const float d = 1.0f - diag;
        attnterm[b * T_ + t0 + M] = anyRow ? d * d : 0.0f;
    }
}

// ---------------------------------------------------------------------------
// Kernel 3: final deterministic reduction of the 4096 per-row partials.
// out = sum(w*nll)/sum(w) + mean(attnterm)
// ---------------------------------------------------------------------------
__global__ __launch_bounds__(256) void finalize_kernel(
    const float* __restrict__ nllw, const float* __restrict__ wsum,
    const float* __restrict__ attnterm, float* __restrict__ out) {
    const int tid = threadIdx.x;
    float s0 = 0.0f, s1 = 0.0f, s2 = 0.0f;
    for (int i = tid; i < NROWS; i += 256) {
        s0 += nllw[i];
        s1 += wsum[i];
        s2 += attnterm[i];
    }
#pragma unroll
    for (int off = 16; off > 0; off >>= 1) {
        s0 += __shfl_xor(s0, off);
        s1 += __shfl_xor(s1, off);
        s2 += __shfl_xor(s2, off);
    }
    __shared__ float a0[8], a1[8], a2[8];
    const int wave = tid >> 5, lane = tid & 31;
    if (lane == 0) { a0[wave] = s0; a1[wave] = s1; a2[wave] = s2; }
    __syncthreads();
    if (tid == 0) {
        for (int w = 1; w < 8; ++w) { s0 += a0[w]; s1 += a1[w]; s2 += a2[w]; }
        out[0] = s0 / s1 + s2 / (float)NROWS;
    }
}

extern "C" void kernel_launch(void* const* d_in, const int* in_sizes, int n_in,
                              void* d_out, int out_size, void* d_ws, size_t ws_size,
                              hipStream_t stream) {
    const float* logits        = (const float*)d_in[0];
    const int* targets         = (const int*)d_in[1];
    const float* attns         = (const float*)d_in[2];
    const unsigned char* assoc = (const unsigned char*)d_in[3];   // bool mask, 1 byte/elem

    // workspace layout: gens[4096] (int) | nllw[4096] | wsum[4096] | attnterm[4096]
    float* ws = (float*)d_ws;
    int* gens       = (int*)ws;
    float* nllw     = ws + NROWS;
    float* wsum     = ws + 2 * NROWS;
    float* attnterm = ws + 3 * NROWS;

    rowpass_kernel<<<NROWS, 256, 0, stream>>>(logits, targets, gens, nllw, wsum);
    attn_wmma_kernel<<<NROWS / 16, 32, 0, stream>>>(attns, assoc, gens, attnterm);
    finalize_kernel<<<1, 256, 0, stream>>>(nllw, wsum, attnterm, (float*)d_out);
}